// SimpleCarDynamics_1039382085786
// MI455X (gfx1250) — compile-verified
//
#include <hip/hip_runtime.h>

typedef float v2f __attribute__((ext_vector_type(2)));
typedef float v8f __attribute__((ext_vector_type(8)));

#define WB_INV   (1.0f / 0.29f)
#define T2W      17.0f
#define SMAX     0.488f
#define DT       0.02f
#define DT_INV   50.0f
#define MAP_ORG  (-25.6f)
#define MAP_IRES 20.0f
#define ROWS     68            // padded LDS row stride in floats (bank-conflict free)
#define CROWS    66            // control panel row stride in float2 (bank-conflict free)
#define NW       2             // waves per block

static __device__ __forceinline__ void lds_fence() {
    // gfx1250 split wait counter for LDS; memory clobber stops reordering.
    asm volatile("s_wait_dscnt 0x0" ::: "memory");
}

// Inclusive prefix-sum along horizon (64) for 16 trajectories via WMMA:
// out = IN(16x64) @ U(64x64), U[k][h] = (k <= h).
// U's block structure: chunks below the diagonal tile are all-ones (folded into
// a running carry accumulator), above are zero (skipped), and the 4 diagonal
// chunks use lane-constant triangular B (identical for every tile).
__device__ __forceinline__ void wmma_cumsum(const float* lds, int lane,
                                            const v2f Bdiag[4], v2f Bones,
                                            v8f acc[4]) {
    const int mA   = lane & 15;          // A layout: M = lane%16
    const int koff = (lane >> 4) << 1;   // A/B layout: K offset 0 or 2 per half-wave
    const float* row = lds + mA * ROWS + koff;
    v2f A[16];
#pragma unroll
    for (int kc = 0; kc < 16; ++kc) {
        A[kc].x = row[4 * kc + 0];
        A[kc].y = row[4 * kc + 1];
    }
    v8f carry = {};                      // sum of all K below current tile, bcast over N
#pragma unroll
    for (int n = 0; n < 4; ++n) {
        v8f d = carry;
#pragma unroll
        for (int j = 0; j < 4; ++j)
            d = __builtin_amdgcn_wmma_f32_16x16x4_f32(
                    false, A[4 * n + j], false, Bdiag[j], (short)0, d, false, false);
        acc[n] = d;
        if (n < 3) {
#pragma unroll
            for (int j = 0; j < 4; ++j)
                carry = __builtin_amdgcn_wmma_f32_16x16x4_f32(
                        false, A[4 * n + j], false, Bones, (short)0, carry, false, false);
        }
    }
}

__global__ __launch_bounds__(NW * 32)
void car_rollout_kernel(const float* __restrict__ x0,
                        const float* __restrict__ controls,
                        const float* __restrict__ map_data,
                        float* __restrict__ out,
                        int n_traj) {
    // Per wave: 16 trajectories; 5 staging panels (wz, dScy, dSsy, cy, sy) plus
    // an LDS-resident copy of this wave's control slice (async-copied once).
    __shared__ float  smem[NW][5][16 * ROWS];
    __shared__ float2 cbuf[NW][16 * CROWS];

    const int lane = threadIdx.x & 31;
    const int wave = threadIdx.x >> 5;
    const int T0   = blockIdx.x * (NW * 16) + wave * 16;
    if (T0 >= n_traj) return;                        // wave-uniform (EXEC stays full)

    float*  B0 = smem[wave][0];   // wz
    float*  B1 = smem[wave][1];   // dS*cos(yaw)  -> later roll
    float*  B2 = smem[wave][2];   // dS*sin(yaw)  -> later pitch
    float*  B3 = smem[wave][3];   // cos(yaw)
    float*  B4 = smem[wave][4];   // sin(yaw)
    float2* CB = cbuf[wave];      // controls [M][h], row stride CROWS

    // ---- async copy: 16 rows x 512B of controls -> LDS (ASYNCcnt-tracked) ----
    {
        const unsigned lbase = (unsigned)(uintptr_t)CB;
        const unsigned long long gbase =
            (unsigned long long)(uintptr_t)(controls) + (unsigned long long)T0 * 512ull;
#pragma unroll
        for (int i = 0; i < 16; ++i) {
            unsigned           ldsa = lbase + (unsigned)(i * (CROWS * 8) + lane * 16);
            unsigned long long ga   = gbase + (unsigned long long)(i * 512 + lane * 16);
            asm volatile("global_load_async_to_lds_b128 %0, %1, off"
                         :: "v"(ldsa), "v"(ga) : "memory");
        }
        asm volatile("s_wait_asynccnt 0x0" ::: "memory");
    }

    const int   env   = T0 >> 13;                    // 8192 samples per env
    const float x0x   = x0[env * 14 + 0];
    const float x0y   = x0[env * 14 + 1];
    const float x0yaw = x0[env * 14 + 5];
    const float wx0   = x0[env * 14 + 9];
    const float wy0   = x0[env * 14 + 10];

    const int mC   = (lane >> 4) << 3;               // C/D layout: M = r + (lane/16)*8
    const int nC   = lane & 15;                      //            N = lane%16
    const int koff = (lane >> 4) << 1;               // A/B layout K offset (0 or 2)

    // Lane-constant triangular B patterns (8 compares, once).
    v2f Bdiag[4];
#pragma unroll
    for (int j = 0; j < 4; ++j) {
        Bdiag[j].x = (4 * j + koff     <= nC) ? 1.0f : 0.0f;
        Bdiag[j].y = (4 * j + koff + 1 <= nC) ? 1.0f : 0.0f;
    }
    v2f Bones; Bones.x = 1.0f; Bones.y = 1.0f;

    // ---------- stage 1: wz = vx * tan(steer*SMAX)/WB  -> B0 ----------
#pragma unroll
    for (int n = 0; n < 4; ++n) {
#pragma unroll
        for (int r = 0; r < 8; ++r) {
            const int M = r + mC, h = 16 * n + nC;
            const float2 c = CB[M * CROWS + h];
            const float vx = c.x * T2W;
            B0[M * ROWS + h] = vx * tanf(c.y * SMAX) * WB_INV;
        }
    }
    lds_fence();

    // ---------- stage 2: yaw = x0yaw + DT * cumsum(wz) ----------
    v8f yawT[4];
    wmma_cumsum(B0, lane, Bdiag, Bones, yawT);

    // ---------- stage 3: sincos once; stage panels for later reuse ----------
#pragma unroll
    for (int n = 0; n < 4; ++n) {
#pragma unroll
        for (int r = 0; r < 8; ++r) {
            const int M = r + mC, h = 16 * n + nC;
            const float2 c  = CB[M * CROWS + h];
            const float dS  = c.x * T2W * DT;
            const float yw  = x0yaw + DT * yawT[n][r];
            yawT[n][r] = yw;                          // keep yaw in regs for output
            float sy, cy;
            sincosf(yw, &sy, &cy);
            B1[M * ROWS + h] = dS * cy;
            B2[M * ROWS + h] = dS * sy;
            B3[M * ROWS + h] = cy;
            B4[M * ROWS + h] = sy;
        }
    }
    lds_fence();

    v8f xT[4], yT[4];
    wmma_cumsum(B1, lane, Bdiag, Bones, xT);
    wmma_cumsum(B2, lane, Bdiag, Bones, yT);
#pragma unroll
    for (int n = 0; n < 4; ++n)
#pragma unroll
        for (int r = 0; r < 8; ++r) {
            xT[n][r] = x0x + xT[n][r];
            yT[n][r] = x0y + yT[n][r];
        }
    lds_fence();   // cumsum A-loads complete before B1/B2 are reused below

    const float4* mp = (const float4*)map_data;

    // ---------- stage 4+5: map gather, roll/pitch, diff, output (tile by tile) ----------
#pragma unroll
    for (int n = 0; n < 4; ++n) {
        float zv[8], roll[8], pitch[8];
#pragma unroll
        for (int r = 0; r < 8; ++r) {
            const int M = r + mC, h = 16 * n + nC;
            const float X = xT[n][r];
            const float Y = yT[n][r];

            float fx = fminf(fmaxf((X - MAP_ORG) * MAP_IRES, 0.0f), 1022.999f);
            float fy = fminf(fmaxf((Y - MAP_ORG) * MAP_IRES, 0.0f), 1022.999f);
            const int ix = (int)fx;
            const int iy = (int)fy;
            const float wxf = fx - (float)ix;
            const float wyf = fy - (float)iy;

            const float4 v00 = mp[ix * 1024 + iy];
            const float4 v10 = mp[(ix + 1) * 1024 + iy];
            const float4 v01 = mp[ix * 1024 + iy + 1];
            const float4 v11 = mp[(ix + 1) * 1024 + iy + 1];

            const float w00 = (1.0f - wxf) * (1.0f - wyf);
            const float w10 = wxf * (1.0f - wyf);
            const float w01 = (1.0f - wxf) * wyf;
            const float w11 = wxf * wyf;

            const float zz = w00 * v00.x + w10 * v10.x + w01 * v01.x + w11 * v11.x;
            const float nx = w00 * v00.y + w10 * v10.y + w01 * v01.y + w11 * v11.y;
            const float ny = w00 * v00.z + w10 * v10.z + w01 * v01.z + w11 * v11.z;
            const float nz = w00 * v00.w + w10 * v10.w + w01 * v01.w + w11 * v11.w;

            const float cy = B3[M * ROWS + h];
            const float sy = B4[M * ROWS + h];

            // left = cross(normal, heading); forward = cross(left, normal)
            const float lz = nx * sy - ny * cy;
            const float lx = -nz * sy;
            const float ly = nz * cy;
            const float fz = lx * ny - ly * nx;

            zv[r]    = zz;
            roll[r]  = asinf(lz);
            pitch[r] = -asinf(fz);

            B1[M * ROWS + h] = roll[r];
            B2[M * ROWS + h] = pitch[r];
        }
        lds_fence();

#pragma unroll
        for (int r = 0; r < 8; ++r) {
            const int M = r + mC, h = 16 * n + nC;
            float wxv, wyv;
            if (h == 0) {
                wxv = wx0;
                wyv = wy0;
            } else {
                wxv = (roll[r]  - B1[M * ROWS + h - 1]) * DT_INV;
                wyv = (pitch[r] - B2[M * ROWS + h - 1]) * DT_INV;
            }
            const float2 c  = CB[M * CROWS + h];
            const float vx  = c.x * T2W;
            const float wz  = B0[M * ROWS + h];      // staged in stage 1

            float2* o = (float2*)(out + ((size_t)(T0 + M) * 64 + h) * 14);
            o[0] = make_float2(xT[n][r], yT[n][r]);
            o[1] = make_float2(zv[r], roll[r]);
            o[2] = make_float2(pitch[r], yawT[n][r]);
            o[3] = make_float2(vx, 0.0f);
            o[4] = make_float2(0.0f, wxv);
            o[5] = make_float2(wyv, wz);
            o[6] = make_float2(c.x, c.y);
        }
        lds_fence();   // diff loads done before next tile overwrites neighbors
    }
}

extern "C" void kernel_launch(void* const* d_in, const int* in_sizes, int n_in,
                              void* d_out, int out_size, void* d_ws, size_t ws_size,
                              hipStream_t stream) {
    const float* x0       = (const float*)d_in[0];
    const float* controls = (const float*)d_in[1];
    const float* map_data = (const float*)d_in[2];
    float*       out      = (float*)d_out;

    const int n_traj = in_sizes[1] / 128;           // horizon(64) * 2 controls
    const int per_blk = NW * 16;
    const int blocks  = (n_traj + per_blk - 1) / per_blk;
    car_rollout_kernel<<<blocks, NW * 32, 0, stream>>>(x0, controls, map_data, out, n_traj);
}